// vector_quantizer_81501299409479
// MI455X (gfx1250) — compile-verified
//
#include <hip/hip_runtime.h>

typedef float v2f __attribute__((ext_vector_type(2)));
typedef float v8f __attribute__((ext_vector_type(8)));

#define B_SZ     16
#define D_DIM    64
#define L_SZ     2048
#define N_ROWS   (B_SZ * L_SZ)          // 32768 flattened vectors
#define K_CODES  1000
#define CHUNK_C  128                    // codes staged in LDS per chunk
#define P_DIM    (D_DIM / 2)            // 32 float2-pairs per code
#define CP_PAD   132                    // padded float2 stride per pair-row
#define NUM_CC   8                      // 8*128 = 1024 >= 1000
#define OUT_ELEMS (B_SZ * D_DIM * L_SZ) // 2097152
#define LOSS_OFF OUT_ELEMS
#define USE_OFF  (OUT_ELEMS + 1)
#define IDX_OFF  (OUT_ELEMS + 2)

// ---------------------------------------------------------------- init ws
__global__ void vq_init(int* __restrict__ counts, float* __restrict__ sse) {
    int t = blockIdx.x * blockDim.x + threadIdx.x;
    if (t < 1024) counts[t] = 0;
    if (t == 0)   sse[0] = 0.0f;
}

// ------------------------------------------------- fused distances+argmin
// One wave32 owns a 16-row tile of flattened x; codebook chunks staged in LDS
// as contiguous (d,d+1) float2 pairs so each B fragment is ONE ds_load_b64.
__global__ __launch_bounds__(256) void vq_argmin(
        const float* __restrict__ x, const float* __restrict__ emb,
        float* __restrict__ out, int* __restrict__ counts) {
    __shared__ v2f   s_embP[P_DIM * CP_PAD]; // [p = d/2][c] float2 pairs
    __shared__ float s_nrm[CHUNK_C];

    const int tid  = threadIdx.x;
    const int wave = tid >> 5;
    const int lane = tid & 31;
    const int m    = lane & 15;   // row within tile (A layout: lane%16 = M)
    const int hi   = lane >> 4;   // half-wave selects K pair {0,1} vs {2,3}

    const int n0   = blockIdx.x * 128 + wave * 16;  // first flattened row
    const int brow = n0 >> 11;                       // batch (L=2048)
    const int l0   = n0 & 2047;
    const float* px = x + ((size_t)brow * D_DIM) * L_SZ + (l0 + m);

    // ---- preload A tile: a[kk] covers d = 4*kk + 2*hi + {0,1}
    v2f a[16];
    float pn = 0.0f;
#pragma unroll
    for (int kk = 0; kk < 16; ++kk) {
        const int d0 = 4 * kk + 2 * hi;
        float a0 = px[(size_t)d0 * L_SZ];
        float a1 = px[(size_t)(d0 + 1) * L_SZ];
        a[kk].x = a0; a[kk].y = a1;
        pn += a0 * a0 + a1 * a1;
    }
    // full row norms: lane L holds ||row (L&15)||^2
    float rownorm = pn + __shfl_xor(pn, 16, 32);
    // per-lane copies aligned to the C/D layout (lane row = r + 8*hi)
    float xnr[8];
#pragma unroll
    for (int r = 0; r < 8; ++r) xnr[r] = __shfl(rownorm, r + 8 * hi, 32);

    float best[8]; int bidx[8];
#pragma unroll
    for (int r = 0; r < 8; ++r) { best[r] = 3.4e38f; bidx[r] = 0; }

    for (int cc = 0; cc < NUM_CC * CHUNK_C; cc += CHUNK_C) {
        // ---- cooperative stage: emb rows cc..cc+127 as float2 pairs.
        // thread -> (c = i/32, p = i%32): 8B/lane coalesced global reads,
        // LDS b64 writes land on distinct bank pairs.
        for (int i = tid; i < CHUNK_C * P_DIM; i += 256) {
            const int c = i >> 5, p = i & 31;
            const int gc = cc + c;
            v2f v;
            if (gc < K_CODES) {
                v.x = emb[(size_t)gc * D_DIM + 2 * p];
                v.y = emb[(size_t)gc * D_DIM + 2 * p + 1];
            } else {
                v.x = 0.0f; v.y = 0.0f;
            }
            s_embP[p * CP_PAD + c] = v;
        }
        __syncthreads();
        if (tid < CHUNK_C) {
            float s = 0.0f;
            for (int p = 0; p < P_DIM; ++p) {
                v2f v = s_embP[p * CP_PAD + tid];
                s += v.x * v.x + v.y * v.y;
            }
            s_nrm[tid] = (cc + tid < K_CODES) ? s : 3.0e38f; // padded codes lose
        }
        __syncthreads();

        // prefetch next chunk of the codebook into cache (global_prefetch_b8)
        const int nc = cc + CHUNK_C;
        if (nc < K_CODES) {
            const char* p = (const char*)(emb + (size_t)nc * D_DIM);
            const int span = ((K_CODES - nc) < CHUNK_C ? (K_CODES - nc) : CHUNK_C)
                             * D_DIM * 4;
            const int off = tid * 128;
            if (off < span) __builtin_prefetch(p + off, 0, 0);
        }

        // ---- 8 code-tiles of 16; each B fragment is a single ds_load_b64,
        //      then a 16-deep f32 WMMA chain. unroll 2 lets the scheduler
        //      hoist tile kt+1's loads above tile kt's WMMA chain.
#pragma unroll 2
        for (int kt = 0; kt < 8; ++kt) {
            const int ccol = kt * 16 + m;        // B layout: lane%16 = N (code)

            v2f bf[16];
#pragma unroll
            for (int kk = 0; kk < 16; ++kk) {
                bf[kk] = s_embP[(2 * kk + hi) * CP_PAD + ccol];
            }

            v8f acc = {0.f,0.f,0.f,0.f,0.f,0.f,0.f,0.f};
#pragma unroll
            for (int kk = 0; kk < 16; ++kk) {
                acc = __builtin_amdgcn_wmma_f32_16x16x4_f32(
                        false, a[kk], false, bf[kk], (short)0, acc, false, false);
            }

            const float en = s_nrm[ccol];
            const int   cg = cc + ccol;
#pragma unroll
            for (int r = 0; r < 8; ++r) {
                const float dist = xnr[r] + en - 2.0f * acc[r];
                if (dist < best[r]) { best[r] = dist; bidx[r] = cg; }
            }
        }
        __syncthreads();
    }

    // ---- argmin reduction across the 16 lanes sharing each row
#pragma unroll
    for (int r = 0; r < 8; ++r) {
        float d = best[r]; int id = bidx[r];
#pragma unroll
        for (int off = 8; off >= 1; off >>= 1) {   // xor stays within half-wave
            float od = __shfl_xor(d, off, 32);
            int   oi = __shfl_xor(id, off, 32);
            if (od < d || (od == d && oi < id)) { d = od; id = oi; }
        }
        if (m == 0) {
            const int n = n0 + r + 8 * hi;
            out[IDX_OFF + n] = (float)id;          // idx output (float-encoded)
            atomicAdd(&counts[id], 1);
        }
    }
}

// ------------------------------ gather winning codes, transpose-out, SSE
__global__ __launch_bounds__(256) void vq_gather(
        const float* __restrict__ x, const float* __restrict__ emb,
        float* __restrict__ out, float* __restrict__ sse) {
    __shared__ float red[256];
    const int bd = blockIdx.x;              // b*64 + d plane
    const int b  = bd >> 6, d = bd & 63;
    const float* idxf = out + IDX_OFF + b * L_SZ;
    const size_t base = (size_t)bd * L_SZ;  // [b][d][l] layout == bd*2048 + l

    float local = 0.0f;
    for (int l = threadIdx.x; l < L_SZ; l += 256) {
        const int   id = (int)idxf[l];
        const float q  = emb[(size_t)id * D_DIM + d];
        const float xv = x[base + l];
        out[base + l] = q;                  // straight-through forward == q
        const float diff = q - xv;
        local += diff * diff;
    }
    red[threadIdx.x] = local;
    __syncthreads();
    for (int s = 128; s; s >>= 1) {
        if (threadIdx.x < s) red[threadIdx.x] += red[threadIdx.x + s];
        __syncthreads();
    }
    if (threadIdx.x == 0) atomicAdd(sse, red[0]);
}

// -------------------------------------------------------------- finalize
__global__ void vq_finalize(const int* __restrict__ counts,
                            const float* __restrict__ sse,
                            float* __restrict__ out) {
    __shared__ int red[256];
    const int t = threadIdx.x;
    int c = 0;
    for (int i = t; i < K_CODES; i += 256) c += (counts[i] > 0) ? 1 : 0;
    red[t] = c;
    __syncthreads();
    for (int s = 128; s; s >>= 1) {
        if (t < s) red[t] += red[t + s];
        __syncthreads();
    }
    if (t == 0) {
        // q_loss + 10*e_loss == 11 * mean((q - x)^2) in the forward pass
        out[LOSS_OFF] = 11.0f * sse[0] / (float)OUT_ELEMS;
        out[USE_OFF]  = (float)red[0];
    }
}

extern "C" void kernel_launch(void* const* d_in, const int* in_sizes, int n_in,
                              void* d_out, int out_size, void* d_ws, size_t ws_size,
                              hipStream_t stream) {
    const float* x   = (const float*)d_in[0];   // [16, 64, 2048]
    const float* emb = (const float*)d_in[1];   // [1000, 64]
    float* out    = (float*)d_out;
    int*   counts = (int*)d_ws;                 // 1024 ints
    float* sse    = (float*)d_ws + 1024;        // 1 float

    vq_init    <<<4,            256, 0, stream>>>(counts, sse);
    vq_argmin  <<<N_ROWS / 128, 256, 0, stream>>>(x, emb, out, counts);
    vq_gather  <<<B_SZ * D_DIM, 256, 0, stream>>>(x, emb, out, sse);
    vq_finalize<<<1,            256, 0, stream>>>(counts, sse, out);
}